// SimpleGraphConvolution_21766894256232
// MI455X (gfx1250) — compile-verified
//
#include <hip/hip_runtime.h>

#define N_NODES 50000
#define N_EDGES 800000
#define IN_F    128
#define OUT_F   64

typedef float v2f __attribute__((ext_vector_type(2)));
typedef float v8f __attribute__((ext_vector_type(8)));

// ---------------------------------------------------------------------------
// GEMM: out[n, o] = sum_k x[n,k] * W[o,k] + b[o]
// One wave -> one 16x16 output tile. K=128 swept in 32 steps of
// V_WMMA_F32_16X16X4_F32 (native fp32 matrix op, exact fp32 numerics).
//
// Operand packing (CDNA5 ISA 7.12.2, wave32):
//   A (16x4 f32, 2 VGPRs): lanes 0-15 hold M=lane, K = 0,1 in v0,v1;
//                          lanes 16-31 hold M=lane-16, K = 2,3.
//   B (4x16 f32, 2 VGPRs): lanes 0-15 hold N=lane, K = 0,1 in v0,v1;
//                          lanes 16-31 hold N=lane-16, K = 2,3.
//   C/D (16x16 f32, 8 VGPRs): VGPR v, lanes 0-15 -> M=v, N=lane;
//                              lanes 16-31 -> M=v+8, N=lane-16.
// 50000 = 3125 * 16 exactly, so no partial tiles; waves that fall outside the
// tile range exit as whole waves (waveId is wave-uniform) keeping EXEC all-1s
// for every WMMA.
// ---------------------------------------------------------------------------
__global__ __launch_bounds__(256) void sgc_gemm_wmma(
    const float* __restrict__ x, const float* __restrict__ W,
    const float* __restrict__ bias, float* __restrict__ out)
{
    const int lane   = threadIdx.x & 31;
    const int waveId = blockIdx.x * 8 + (threadIdx.x >> 5);
    const int NTILES = (N_NODES / 16) * (OUT_F / 16);   // 3125 * 4 = 12500
    if (waveId >= NTILES) return;                        // wave-uniform exit

    const int tileM = waveId >> 2;          // node tile   0..3124
    const int tileN = waveId & 3;           // out-f tile  0..3
    const int node0 = tileM * 16;
    const int out0  = tileN * 16;

    const int mn = lane & 15;               // M for A, N for B
    const int kh = (lane >> 4) * 2;         // K half select (0 or 2)

    const float* __restrict__ xrow = x + (size_t)(node0 + mn) * IN_F;
    const float* __restrict__ wrow = W + (size_t)(out0 + mn) * IN_F;

    v8f acc = {};
#pragma unroll
    for (int kb = 0; kb < IN_F; kb += 4) {
        v2f a, b;
        a.x = xrow[kb + kh + 0];
        a.y = xrow[kb + kh + 1];
        b.x = wrow[kb + kh + 0];
        b.y = wrow[kb + kh + 1];
        // (neg_a, A, neg_b, B, c_mod, C, reuse_a, reuse_b)
        acc = __builtin_amdgcn_wmma_f32_16x16x4_f32(
            false, a, false, b, (short)0, acc, false, false);
    }

    const int   n  = lane & 15;
    const float bv = bias[out0 + n];
    const int   m0 = node0 + (lane >> 4) * 8;
#pragma unroll
    for (int v = 0; v < 8; ++v)
        out[(size_t)(m0 + v) * OUT_F + out0 + n] = acc[v] + bv;
}

// ---------------------------------------------------------------------------
// Zero-fill (graph-capture-safe replacement for memset), float4 granularity.
// ---------------------------------------------------------------------------
__global__ __launch_bounds__(256) void sgc_zero_f4(float4* __restrict__ p, int n4)
{
    int i = blockIdx.x * 256 + threadIdx.x;
    if (i < n4) p[i] = make_float4(0.f, 0.f, 0.f, 0.f);
}

// ---------------------------------------------------------------------------
// SpMM scatter: out[row] += edge_attr[e] * in[col]  (64 features).
// 16 threads per edge, float4 gather + 4 device-scope no-return f32 atomics
// (lowers to global_atomic_add_f32; RMW resolves in L2 where the whole
// feature buffer is resident).
// ---------------------------------------------------------------------------
__global__ __launch_bounds__(256) void sgc_spmm_scatter(
    const long long* __restrict__ eidx, const float* __restrict__ eattr,
    const float* __restrict__ in, float* __restrict__ out)
{
    const int gid = blockIdx.x * 256 + threadIdx.x;
    const int e   = gid >> 4;
    if (e >= N_EDGES) return;
    const int f = (gid & 15) << 2;

    const long long row = eidx[e];
    const long long col = eidx[N_EDGES + e];
    const float     a   = eattr[e];

    const float4 v = *(const float4*)(in + col * OUT_F + f);
    float* __restrict__ dst = out + row * OUT_F + f;

    __hip_atomic_fetch_add(dst + 0, v.x * a, __ATOMIC_RELAXED, __HIP_MEMORY_SCOPE_AGENT);
    __hip_atomic_fetch_add(dst + 1, v.y * a, __ATOMIC_RELAXED, __HIP_MEMORY_SCOPE_AGENT);
    __hip_atomic_fetch_add(dst + 2, v.z * a, __ATOMIC_RELAXED, __HIP_MEMORY_SCOPE_AGENT);
    __hip_atomic_fetch_add(dst + 3, v.w * a, __ATOMIC_RELAXED, __HIP_MEMORY_SCOPE_AGENT);
}

extern "C" void kernel_launch(void* const* d_in, const int* in_sizes, int n_in,
                              void* d_out, int out_size, void* d_ws, size_t ws_size,
                              hipStream_t stream)
{
    const float*     x     = (const float*)d_in[0];
    const long long* eidx  = (const long long*)d_in[1];
    const float*     eattr = (const float*)d_in[2];
    const float*     W     = (const float*)d_in[3];
    const float*     bias  = (const float*)d_in[4];
    float*           out   = (float*)d_out;

    const size_t FEAT = (size_t)N_NODES * OUT_F;   // 3.2M floats
    float* bufA = (float*)d_ws;
    float* bufB = bufA + FEAT;                     // needs 25.6 MB of ws

    const int zeroBlocks = (int)((FEAT / 4 + 255) / 256);      // 3125
    const int gemmBlocks = ((N_NODES / 16) * (OUT_F / 16) + 7) / 8; // 1563
    const int spmmBlocks = (N_EDGES * 16) / 256;               // 50000

    // Linear: bufA = x @ W^T + b   (WMMA fp32)
    sgc_gemm_wmma<<<gemmBlocks, 256, 0, stream>>>(x, W, bias, bufA);

    // Round 1: bufB = spmm(bufA)
    sgc_zero_f4<<<zeroBlocks, 256, 0, stream>>>((float4*)bufB, (int)(FEAT / 4));
    sgc_spmm_scatter<<<spmmBlocks, 256, 0, stream>>>(eidx, eattr, bufA, bufB);

    // Round 2: bufA = spmm(bufB)
    sgc_zero_f4<<<zeroBlocks, 256, 0, stream>>>((float4*)bufA, (int)(FEAT / 4));
    sgc_spmm_scatter<<<spmmBlocks, 256, 0, stream>>>(eidx, eattr, bufB, bufA);

    // Round 3: d_out = spmm(bufA)
    sgc_zero_f4<<<zeroBlocks, 256, 0, stream>>>((float4*)out, (int)(FEAT / 4));
    sgc_spmm_scatter<<<spmmBlocks, 256, 0, stream>>>(eidx, eattr, bufA, out);
}